// JointIntegralRegressor_61735859913531
// MI455X (gfx1250) — compile-verified
//
#include <hip/hip_runtime.h>
#include <cmath>

// Problem geometry (fixed by the reference): heatmaps (16,24,64,64,64) f32.
#define NJ       384                 // 16*24 joints
#define VOL      262144              // 64*64*64 elements per (n,j)
#define SPLIT    4                   // blocks per (n,j) volume -> 1536 blocks
#define THREADS  256                 // 8 wave32s
#define F4_TOTAL (VOL / 4)           // 65536 float4 per (n,j)
#define F4_BLK   (F4_TOTAL / SPLIT)  // 16384 float4 per block
#define ITERS    (F4_BLK / THREADS)  // 64 float4 per thread
#define PIPE     8                   // async global->LDS pipeline depth

#define AS1 __attribute__((address_space(1)))
#define AS3 __attribute__((address_space(3)))

// Matches the builtin's parameter pointee exactly (GNU vector, not ext_vector):
// 'cannot initialize a parameter of type __vector_size__(4*sizeof(int)) int __device__ *'
typedef int v4i __attribute__((vector_size(16)));

// CDNA5 async copy path (device pass only; host pass falls back cleanly).
#if __has_builtin(__builtin_amdgcn_global_load_async_to_lds_b128)
#  define HAVE_ASYNC 1
#endif

#ifdef HAVE_ASYNC
#  if __has_builtin(__builtin_amdgcn_s_wait_asynccnt)
#    define WAIT_ASYNC(n) __builtin_amdgcn_s_wait_asynccnt(n)
#  else
#    define WAIT_ASYNC(n) asm volatile("s_wait_asynccnt %0" :: "n"(n) : "memory")
#  endif
// global -> LDS async copy of one 16B element per lane
#  define ASYNC_CP_B128(gp, lp)                                            \
     __builtin_amdgcn_global_load_async_to_lds_b128(                       \
         (AS1 v4i*)(gp), (AS3 v4i*)(lp), 0, 0)
#endif

// Merge two online-softmax partial states: (m, S, Sx, Sy, Sz).
__device__ __forceinline__ void softmax_merge(
    float& m1, float& s1, float& x1, float& y1, float& z1,
    float m2, float s2, float x2, float y2, float z2) {
  float M = fmaxf(m1, m2);
  float a = __expf(m1 - M);
  float b = __expf(m2 - M);
  m1 = M;
  s1 = fmaf(a, s1, b * s2);
  x1 = fmaf(a, x1, b * x2);
  y1 = fmaf(a, y1, b * y2);
  z1 = fmaf(a, z1, b * z2);
}

// Accumulate one float4 (4 consecutive W entries sharing one h,d).
// sxd accumulates only the intra-float4 w offsets (0..3); the constant per-
// thread w0 term is applied once at the end as w0*S + sxd.
__device__ __forceinline__ void accum4(
    float4 v, float fh, float fd,
    float& m, float& S, float& sxd, float& sy, float& sz) {
  float vmax = fmaxf(fmaxf(v.x, v.y), fmaxf(v.z, v.w));
  if (vmax > m) {                      // rare: O(log n) times per thread
    float r = __expf(m - vmax);        // exp(-inf)=0 handles the first iter
    S *= r; sxd *= r; sy *= r; sz *= r;
    m = vmax;
  }
  float e0 = __expf(v.x - m);
  float e1 = __expf(v.y - m);
  float e2 = __expf(v.z - m);
  float e3 = __expf(v.w - m);
  float es = (e0 + e1) + (e2 + e3);
  float dt = fmaf(3.f, e3, fmaf(2.f, e2, e1));
  S   += es;
  sxd += dt;
  sy   = fmaf(es, fh, sy);
  sz   = fmaf(es, fd, sz);
}

__global__ __launch_bounds__(THREADS)
void jir_pass1(const float* __restrict__ hm, float* __restrict__ part) {
  const int b   = (int)blockIdx.x;       // 0 .. NJ*SPLIT-1
  const int nj  = b >> 2;                // b / SPLIT
  const int seg = b & (SPLIT - 1);
  const int t   = (int)threadIdx.x;

  const float4* gf4 = (const float4*)hm + (size_t)nj * F4_TOTAL;
  const int     f4b = seg * F4_BLK;

  // element index e0 = seg*65536 + 1024*k + 4*t  (multiple-of-4 along W)
  //   w0 = (4t) & 63                      (constant per thread)
  //   h  = (t>>4) + ((16k) & 63)          (no carry: <=15 + <=48)
  //   d  = 16*seg + (k>>2)
  const float w0  = (float)((4 * t) & 63);
  const int   hbb = t >> 4;
  const int   dbb = seg << 4;

  float m = -__builtin_inff();
  float S = 0.f, sxd = 0.f, sy = 0.f, sz = 0.f;

#ifdef HAVE_ASYNC
  // Per-wave LDS ring: 8 waves x PIPE stages x 32 lanes x 16B = 32 KB.
  // Each lane async-copies its own 16B and reads back the same slot, so
  // only s_wait_asynccnt ordering is needed (no cross-wave barriers).
  __shared__ float4 stage[THREADS / 32][PIPE][32];
  const int wv = t >> 5, ln = t & 31;

  #pragma unroll
  for (int k = 0; k < PIPE; ++k) {
    ASYNC_CP_B128(gf4 + f4b + k * THREADS + t, &stage[wv][k][ln]);
  }
  for (int k = 0; k < ITERS - PIPE; ++k) {
    WAIT_ASYNC(PIPE - 1);                       // op k completed (in-order)
    asm volatile("" ::: "memory");
    float4 v = stage[wv][k & (PIPE - 1)][ln];   // ds_load_b128
    float fh = (float)(hbb + ((k << 4) & 63));
    float fd = (float)(dbb + (k >> 2));
    accum4(v, fh, fd, m, S, sxd, sy, sz);
    asm volatile("s_wait_dscnt 0x0" ::: "memory");  // read before refill
    ASYNC_CP_B128(gf4 + f4b + (k + PIPE) * THREADS + t,
                  &stage[wv][k & (PIPE - 1)][ln]);
  }
  WAIT_ASYNC(0);                                // drain the tail
  asm volatile("" ::: "memory");
  #pragma unroll
  for (int k = ITERS - PIPE; k < ITERS; ++k) {
    float4 v = stage[wv][k & (PIPE - 1)][ln];
    float fh = (float)(hbb + ((k << 4) & 63));
    float fd = (float)(dbb + (k >> 2));
    accum4(v, fh, fd, m, S, sxd, sy, sz);
  }
#else
  // Fallback: direct b128 streaming loads + gfx1250 global_prefetch_b8.
  for (int k = 0; k < ITERS; ++k) {
    if (k + PIPE < ITERS)
      __builtin_prefetch(gf4 + f4b + (k + PIPE) * THREADS + t, 0, 0);
    float4 v = gf4[f4b + k * THREADS + t];
    float fh = (float)(hbb + ((k << 4) & 63));
    float fd = (float)(dbb + (k >> 2));
    accum4(v, fh, fd, m, S, sxd, sy, sz);
  }
#endif

  // Block-level tree reduction of the 5-float online-softmax state.
  __shared__ float rm[THREADS], rs[THREADS], rx[THREADS], ry[THREADS], rz[THREADS];
  float sx = fmaf(w0, S, sxd);
  rm[t] = m; rs[t] = S; rx[t] = sx; ry[t] = sy; rz[t] = sz;
  __syncthreads();
  for (int off = THREADS >> 1; off > 0; off >>= 1) {
    if (t < off) {
      float m1 = rm[t], s1 = rs[t], x1 = rx[t], y1 = ry[t], z1 = rz[t];
      softmax_merge(m1, s1, x1, y1, z1,
                    rm[t + off], rs[t + off], rx[t + off], ry[t + off], rz[t + off]);
      rm[t] = m1; rs[t] = s1; rx[t] = x1; ry[t] = y1; rz[t] = z1;
    }
    __syncthreads();
  }
  if (t == 0) {
    float* p = part + (size_t)b * 8;   // 8-float stride per partial
    p[0] = rm[0]; p[1] = rs[0]; p[2] = rx[0]; p[3] = ry[0]; p[4] = rz[0];
  }
}

__global__ void jir_pass2(const float* __restrict__ part, float* __restrict__ out) {
  int nj = (int)(blockIdx.x * blockDim.x + threadIdx.x);
  if (nj >= NJ) return;
  const float* p = part + (size_t)nj * (SPLIT * 8);
  float m = p[0], s = p[1], x = p[2], y = p[3], z = p[4];
  #pragma unroll
  for (int i = 1; i < SPLIT; ++i) {
    const float* q = p + i * 8;
    softmax_merge(m, s, x, y, z, q[0], q[1], q[2], q[3], q[4]);
  }
  float inv = 1.0f / (64.0f * s);      // expectation / W  (W=H=D=64)
  out[nj * 3 + 0] = fmaf(x, inv, -0.5f);
  out[nj * 3 + 1] = fmaf(y, inv, -0.5f);
  out[nj * 3 + 2] = fmaf(z, inv, -0.5f);
}

extern "C" void kernel_launch(void* const* d_in, const int* in_sizes, int n_in,
                              void* d_out, int out_size, void* d_ws, size_t ws_size,
                              hipStream_t stream) {
  (void)in_sizes; (void)n_in; (void)out_size; (void)ws_size;
  const float* hm   = (const float*)d_in[0];
  float*       part = (float*)d_ws;          // NJ*SPLIT*8 floats = 48 KB
  float*       out  = (float*)d_out;         // (16,24,3) f32

  jir_pass1<<<NJ * SPLIT, THREADS, 0, stream>>>(hm, part);
  jir_pass2<<<(NJ + 127) / 128, 128, 0, stream>>>(part, out);
}